// CapsuleModel_73718818668908
// MI455X (gfx1250) — compile-verified
//
#include <hip/hip_runtime.h>
#include <hip/hip_bf16.h>
#include <math.h>

// ---------------- types for WMMA ----------------
typedef float v2f __attribute__((ext_vector_type(2)));
typedef float v8f __attribute__((ext_vector_type(8)));

// ---------------- model constants ----------------
#define B_SZ   64
#define D_IN   227989LL
#define H_SZ   512
#define NU     8
#define CO     8
#define KK     8
#define L_SZ   253
#define PSIZE  2024          // CO * L_SZ
#define NOUT   4
#define USIZE  64
#define JD     256           // NOUT * USIZE
#define KCHUNK 2048LL
#define NCHUNK 112           // ceil(227989 / 2048)
#define CCHUNK 8             // c-split for s_j (2024 = 8*253)

// =====================================================================
// Kernel 1: fc partial GEMM with V_WMMA_F32_16X16X4_F32
// grid = (32 N-strips, 112 K-chunks), block = 32 (one wave)
// Each wave: full M=64 x N=16 strip -> 4 accumulators over its K chunk.
// 32-bit in-chunk offsets; prefetch hoisted; inner loop unrolled x4.
// =====================================================================
__global__ __launch_bounds__(32) void fc_partial_kernel(
    const float* __restrict__ x, const float* __restrict__ w,
    float* __restrict__ part) {
  const int lane = threadIdx.x;        // 0..31
  const int r    = lane & 15;
  const int hi   = lane >> 4;          // 0 or 1
  const int n0   = blockIdx.x * 16;
  const int chunk= blockIdx.y;

  const long long k0 = (long long)chunk * KCHUNK;
  const int len  = (int)(((k0 + KCHUNK) > D_IN) ? (D_IN - k0) : KCHUNK);
  const int len4 = len & ~3;

  // per-lane base pointers at this chunk origin (+2*hi for the K-pair split)
  const float* __restrict__ wp  = w + (long long)(n0 + r) * D_IN + k0 + 2 * hi;
  const float* __restrict__ xp0 = x + (long long)(r     ) * D_IN + k0 + 2 * hi;
  const float* __restrict__ xp1 = x + (long long)(r + 16) * D_IN + k0 + 2 * hi;
  const float* __restrict__ xp2 = x + (long long)(r + 32) * D_IN + k0 + 2 * hi;
  const float* __restrict__ xp3 = x + (long long)(r + 48) * D_IN + k0 + 2 * hi;

  v8f acc0 = {}, acc1 = {}, acc2 = {}, acc3 = {};

  for (int kb = 0; kb < len4; kb += 256) {
    __builtin_prefetch((const void*)(wp + kb + 2048), 0, 1);
    const int ke = (kb + 256 < len4) ? (kb + 256) : len4;
    #pragma unroll 4
    for (int kk = kb; kk < ke; kk += 4) {
      v2f bf; bf.x = wp [kk]; bf.y = wp [kk + 1];
      v2f a0; a0.x = xp0[kk]; a0.y = xp0[kk + 1];
      v2f a1; a1.x = xp1[kk]; a1.y = xp1[kk + 1];
      v2f a2; a2.x = xp2[kk]; a2.y = xp2[kk + 1];
      v2f a3; a3.x = xp3[kk]; a3.y = xp3[kk + 1];
      acc0 = __builtin_amdgcn_wmma_f32_16x16x4_f32(false, a0, false, bf, (short)0, acc0, false, false);
      acc1 = __builtin_amdgcn_wmma_f32_16x16x4_f32(false, a1, false, bf, (short)0, acc1, false, false);
      acc2 = __builtin_amdgcn_wmma_f32_16x16x4_f32(false, a2, false, bf, (short)0, acc2, false, false);
      acc3 = __builtin_amdgcn_wmma_f32_16x16x4_f32(false, a3, false, bf, (short)0, acc3, false, false);
    }
  }
  if (len4 < len) {                    // K tail (1..3 elems), zero-padded
    const int ka = len4;               // pointers already carry +2*hi
    const bool v0 = (ka + 2 * hi     < len);
    const bool v1 = (ka + 2 * hi + 1 < len);
    v2f bf; bf.x = v0 ? wp [ka] : 0.f; bf.y = v1 ? wp [ka + 1] : 0.f;
    v2f a0; a0.x = v0 ? xp0[ka] : 0.f; a0.y = v1 ? xp0[ka + 1] : 0.f;
    v2f a1; a1.x = v0 ? xp1[ka] : 0.f; a1.y = v1 ? xp1[ka + 1] : 0.f;
    v2f a2; a2.x = v0 ? xp2[ka] : 0.f; a2.y = v1 ? xp2[ka + 1] : 0.f;
    v2f a3; a3.x = v0 ? xp3[ka] : 0.f; a3.y = v1 ? xp3[ka + 1] : 0.f;
    acc0 = __builtin_amdgcn_wmma_f32_16x16x4_f32(false, a0, false, bf, (short)0, acc0, false, false);
    acc1 = __builtin_amdgcn_wmma_f32_16x16x4_f32(false, a1, false, bf, (short)0, acc1, false, false);
    acc2 = __builtin_amdgcn_wmma_f32_16x16x4_f32(false, a2, false, bf, (short)0, acc2, false, false);
    acc3 = __builtin_amdgcn_wmma_f32_16x16x4_f32(false, a3, false, bf, (short)0, acc3, false, false);
  }

  // C/D layout: VGPR v -> M = 16*t + v + 8*hi, N = n0 + r
  float* base = part + (long long)chunk * 64 * 512;
  const int n = n0 + r;
  #pragma unroll
  for (int v = 0; v < 8; ++v) {
    base[( 0 + v + 8 * hi) * 512 + n] = acc0[v];
    base[(16 + v + 8 * hi) * 512 + n] = acc1[v];
    base[(32 + v + 8 * hi) * 512 + n] = acc2[v];
    base[(48 + v + 8 * hi) * 512 + n] = acc3[v];
  }
}

// =====================================================================
// Kernel 2: reduce K-chunk partials + bias + ReLU -> sdae (output #2)
// =====================================================================
__global__ __launch_bounds__(256) void fc_reduce_kernel(
    const float* __restrict__ part, const float* __restrict__ fc_b,
    float* __restrict__ sdae) {
  const int idx = blockIdx.x * 256 + threadIdx.x;   // 0..32767
  const int n = idx & 511;
  const int m = idx >> 9;
  float s = 0.f;
  #pragma unroll 4
  for (int c = 0; c < NCHUNK; ++c)
    s += part[((long long)c * 64 + m) * 512 + n];
  s += fc_b[n];
  sdae[idx] = fmaxf(s, 0.f);
}

// =====================================================================
// Kernel 3: conv (stride 2, K=8) + bias + squash over the 2024-dim
// grid = (B, NU), block = 256
// =====================================================================
__global__ __launch_bounds__(256) void conv_squash_kernel(
    const float* __restrict__ sdae, const float* __restrict__ conv_w,
    const float* __restrict__ conv_b, float* __restrict__ u) {
  __shared__ float s_in[H_SZ];
  __shared__ float s_val[PSIZE];
  __shared__ float s_red[256];
  const int b = blockIdx.x, nu = blockIdx.y, t = threadIdx.x;

  for (int i = t; i < H_SZ; i += 256) s_in[i] = sdae[b * H_SZ + i];
  __syncthreads();

  float ssq = 0.f;
  for (int e = t; e < PSIZE; e += 256) {
    const int co = e / L_SZ;
    const int l  = e - co * L_SZ;
    const int ch = nu * CO + co;
    const float* wk = conv_w + ch * KK;
    float acc = conv_b[ch];
    const int base = 2 * l;
    #pragma unroll
    for (int k = 0; k < KK; ++k) acc += s_in[base + k] * wk[k];
    s_val[e] = acc;
    ssq += acc * acc;
  }
  s_red[t] = ssq; __syncthreads();
  for (int o = 128; o > 0; o >>= 1) { if (t < o) s_red[t] += s_red[t + o]; __syncthreads(); }
  const float magsq = s_red[0];
  const float scale = sqrtf(magsq) / (1.f + magsq);   // = magsq/(1+magsq)/mag
  for (int e = t; e < PSIZE; e += 256)
    u[((long long)b * NU + nu) * PSIZE + e] = s_val[e] * scale;
}

// =====================================================================
// Kernel 4: u_hat[b,c,j,d] = sum_i W[c,j,d,i] * u[b,i,c]
// grid = 2024 (c), block = 256 (t = j*64+d); W_dig row staged in LDS and
// reused for all 64 batch elements -> W_dig streamed from L2/HBM once.
// =====================================================================
__global__ __launch_bounds__(256) void uhat_kernel(
    const float* __restrict__ u, const float* __restrict__ W,
    float* __restrict__ uhat) {
  const int c = blockIdx.x, t = threadIdx.x;
  __shared__ float sW[JD * NU];       // 2048: W[c, j, d, i] as [t][i]
  __shared__ float su[B_SZ * NU];     // 512:  su[b][i] = u[b, i, c]
  for (int q = t; q < JD * NU; q += 256) sW[q] = W[(long long)c * JD * NU + q];
  for (int q = t; q < B_SZ * NU; q += 256) {
    const int b = q >> 3, i = q & 7;
    su[q] = u[((long long)b * NU + i) * PSIZE + c];
  }
  __syncthreads();
  float wreg[NU];
  #pragma unroll
  for (int i = 0; i < NU; ++i) wreg[i] = sW[t * NU + i];
  for (int b = 0; b < B_SZ; ++b) {
    float acc = 0.f;
    #pragma unroll
    for (int i = 0; i < NU; ++i) acc += wreg[i] * su[b * NU + i];
    uhat[((long long)b * PSIZE + c) * JD + t] = acc;
  }
}

// =====================================================================
// Routing kernels
// =====================================================================
__global__ __launch_bounds__(256) void softmax_kernel(
    const float* __restrict__ bij, float* __restrict__ cij) {
  const int j = blockIdx.x, t = threadIdx.x;
  __shared__ float red[256];
  float m = -1e30f;
  for (int c = t; c < PSIZE; c += 256) m = fmaxf(m, bij[c * 4 + j]);
  red[t] = m; __syncthreads();
  for (int o = 128; o > 0; o >>= 1) { if (t < o) red[t] = fmaxf(red[t], red[t + o]); __syncthreads(); }
  m = red[0]; __syncthreads();
  float s = 0.f;
  for (int c = t; c < PSIZE; c += 256) s += __expf(bij[c * 4 + j] - m);
  red[t] = s; __syncthreads();
  for (int o = 128; o > 0; o >>= 1) { if (t < o) red[t] += red[t + o]; __syncthreads(); }
  const float inv = 1.f / red[0];
  for (int c = t; c < PSIZE; c += 256) cij[c * 4 + j] = __expf(bij[c * 4 + j] - m) * inv;
}

// partial s_j over a c-chunk: grid=(4, 64, 8), block=64
__global__ __launch_bounds__(64) void sj_partial_kernel(
    const float* __restrict__ uhat, const float* __restrict__ cij,
    float* __restrict__ sjp) {
  const int j = blockIdx.x, b = blockIdx.y, cc = blockIdx.z, d = threadIdx.x;
  const int c0 = cc * (PSIZE / CCHUNK);           // 253 per chunk
  const int c1 = c0 + (PSIZE / CCHUNK);
  const float* up = uhat + (long long)b * PSIZE * JD + j * USIZE + d;
  float s = 0.f;
  for (int c = c0; c < c1; ++c) s += cij[c * 4 + j] * up[(long long)c * JD];
  sjp[(((long long)cc * B_SZ + b) * NOUT + j) * USIZE + d] = s;
}

// fold c-chunks + squash over j: grid=64 (b), block=64 (d)
__global__ __launch_bounds__(64) void squashv_kernel(
    const float* __restrict__ sjp, float* __restrict__ vj) {
  const int b = blockIdx.x, d = threadIdx.x;
  float s[NOUT];
  #pragma unroll
  for (int j = 0; j < NOUT; ++j) {
    float a = 0.f;
    #pragma unroll
    for (int cc = 0; cc < CCHUNK; ++cc)
      a += sjp[(((long long)cc * B_SZ + b) * NOUT + j) * USIZE + d];
    s[j] = a;
  }
  const float magsq = s[0]*s[0] + s[1]*s[1] + s[2]*s[2] + s[3]*s[3];
  const float scale = sqrtf(magsq) / (1.f + magsq);
  #pragma unroll
  for (int j = 0; j < NOUT; ++j)
    vj[(b * NOUT + j) * USIZE + d] = s[j] * scale;
}

// b_ij[c,j] += mean_b sum_d uhat[b,c,j,d]*v[b,j,d]   grid=(4,2024), block=64
__global__ __launch_bounds__(64) void update_kernel(
    const float* __restrict__ uhat, const float* __restrict__ vj,
    float* __restrict__ bij) {
  const int j = blockIdx.x, c = blockIdx.y, d = threadIdx.x;
  float a = 0.f;
  for (int b = 0; b < B_SZ; ++b)
    a += uhat[((long long)b * PSIZE + c) * JD + j * USIZE + d] *
         vj[(b * NOUT + j) * USIZE + d];
  __shared__ float red[64];
  red[d] = a; __syncthreads();
  for (int o = 32; o > 0; o >>= 1) { if (d < o) red[d] += red[d + o]; __syncthreads(); }
  if (d == 0) bij[c * 4 + j] += red[0] * (1.f / 64.f);
}

// =====================================================================
// Kernel: final MLP, 256 -> 128 -> 64 -> 2, LeakyReLU(0.01)
// grid = 64 (b), block = 128
// =====================================================================
__global__ __launch_bounds__(128) void mlp_kernel(
    const float* __restrict__ vj,
    const float* __restrict__ w1, const float* __restrict__ b1,
    const float* __restrict__ w2, const float* __restrict__ b2,
    const float* __restrict__ w3, const float* __restrict__ b3,
    float* __restrict__ logits) {
  const int b = blockIdx.x, t = threadIdx.x;
  __shared__ float caps[256], h1[128], h2[64];
  caps[t]       = vj[b * 256 + t];
  caps[t + 128] = vj[b * 256 + t + 128];
  __syncthreads();
  {
    float a = b1[t];
    for (int k = 0; k < 256; ++k) a += w1[t * 256 + k] * caps[k];
    h1[t] = (a >= 0.f) ? a : 0.01f * a;
  }
  __syncthreads();
  if (t < 64) {
    float a = b2[t];
    for (int k = 0; k < 128; ++k) a += w2[t * 128 + k] * h1[k];
    h2[t] = (a >= 0.f) ? a : 0.01f * a;
  }
  __syncthreads();
  if (t < 2) {
    float a = b3[t];
    for (int k = 0; k < 64; ++k) a += w3[t * 64 + k] * h2[k];
    logits[b * 2 + t] = a;
  }
}

// =====================================================================
// Launch
// =====================================================================
extern "C" void kernel_launch(void* const* d_in, const int* in_sizes, int n_in,
                              void* d_out, int out_size, void* d_ws, size_t ws_size,
                              hipStream_t stream) {
  const float* x      = (const float*)d_in[0];
  const float* fc_w   = (const float*)d_in[1];
  const float* fc_b   = (const float*)d_in[2];
  const float* conv_w = (const float*)d_in[3];
  const float* conv_b = (const float*)d_in[4];
  const float* W_dig  = (const float*)d_in[5];
  const float* f2w1   = (const float*)d_in[6];
  const float* f2b1   = (const float*)d_in[7];
  const float* f2w2   = (const float*)d_in[8];
  const float* f2b2   = (const float*)d_in[9];
  const float* f2w3   = (const float*)d_in[10];
  const float* f2b3   = (const float*)d_in[11];

  float* out    = (float*)d_out;
  float* logits = out;            // [64,2]
  float* sdae   = out + 128;      // [64,512]

  // workspace layout (floats)
  float* ws   = (float*)d_ws;
  float* uhat = ws;                                   // 64*2024*256 = 33,161,216
  float* part = uhat + (long long)B_SZ * PSIZE * JD;  // 112*64*512 =  3,670,016
  float* u    = part + (long long)NCHUNK * 64 * 512;  // 64*8*2024  =  1,036,288
  float* cij  = u    + (long long)B_SZ * NU * PSIZE;  // 8096
  float* sjp  = cij  + PSIZE * NOUT;                  // 8*64*4*64  = 131,072
  float* vj   = sjp  + (long long)CCHUNK * B_SZ * NOUT * USIZE; // 16384
  float* bij  = vj   + B_SZ * NOUT * USIZE;           // 8096

  // 1) fc GEMM (WMMA f32), K split into 112 chunks for occupancy
  fc_partial_kernel<<<dim3(32, NCHUNK), 32, 0, stream>>>(x, fc_w, part);
  // 2) reduce + bias + ReLU -> sdae (output)
  fc_reduce_kernel<<<(B_SZ * H_SZ) / 256, 256, 0, stream>>>(part, fc_b, sdae);
  // 3) primary capsules: conv + squash
  conv_squash_kernel<<<dim3(B_SZ, NU), 256, 0, stream>>>(sdae, conv_w, conv_b, u);
  // 4) u_hat (132 MB, L2 resident)
  uhat_kernel<<<PSIZE, 256, 0, stream>>>(u, W_dig, uhat);

  // 5) dynamic routing, 3 iterations
  hipMemsetAsync(bij, 0, PSIZE * NOUT * sizeof(float), stream);
  for (int it = 0; it < 3; ++it) {
    softmax_kernel<<<NOUT, 256, 0, stream>>>(bij, cij);
    sj_partial_kernel<<<dim3(NOUT, B_SZ, CCHUNK), 64, 0, stream>>>(uhat, cij, sjp);
    squashv_kernel<<<B_SZ, 64, 0, stream>>>(sjp, vj);
    if (it < 2)
      update_kernel<<<dim3(NOUT, PSIZE), 64, 0, stream>>>(uhat, vj, bij);
  }

  // 6) classifier MLP
  mlp_kernel<<<B_SZ, 128, 0, stream>>>(vj, f2w1, f2b1, f2w2, f2b2, f2w3, f2b3, logits);
}